// GAT_32066225832635
// MI455X (gfx1250) — compile-verified
//
#include <hip/hip_runtime.h>
#include <hip/hip_bf16.h>

#define NN 100000
#define EE 1600000
#define HH 4

typedef __attribute__((ext_vector_type(2))) float v2f;
typedef __attribute__((ext_vector_type(8))) float v8f;

__device__ __forceinline__ void atomAddF(float* p, float v) {
    unsafeAtomicAdd(p, v);   // native global_atomic_add_f32
}

__device__ __forceinline__ void atomMaxF(float* addr, float val) {
    // IEEE float max via integer atomics (works for all sign combos)
    if (val >= 0.0f) atomicMax((int*)addr, __float_as_int(val));
    else             atomicMin((unsigned int*)addr, __float_as_uint(val));
}

// ---------------------------------------------------------------- fill
__global__ void gat_fill_f32(float* __restrict__ p, float v, long long n) {
    long long i = (long long)blockIdx.x * blockDim.x + threadIdx.x;
    if (i < n) p[i] = v;
}

// ------------------------------------------------- pad B [K,F] -> Bp [K,Fp]
__global__ void gat_pad_B(const float* __restrict__ B, float* __restrict__ Bp,
                          int K, int F, int Fp) {
    int t = blockIdx.x * blockDim.x + threadIdx.x;
    if (t >= K * Fp) return;
    int k = t / Fp, f = t - k * Fp;
    Bp[t] = (f < F) ? B[(size_t)k * F + f] : 0.f;
}

// ------------------------------------------------- fold Wd with ar -> vd[k,h]
__global__ void gat_fold_vd(const float* __restrict__ Wd, const float* __restrict__ ar,
                            float* __restrict__ vd, int K, int F, int Dh) {
    int t = blockIdx.x * blockDim.x + threadIdx.x;
    if (t >= K * HH) return;
    int k = t >> 2, h = t & 3;
    const float* wr = Wd + (size_t)k * F + h * Dh;
    const float* av = ar + h * Dh;
    float s = 0.f;
    for (int d = 0; d < Dh; ++d) s += wr[d] * av[d];
    vd[k * HH + h] = s;
}

// ------------------------------------------------- fp32 WMMA GEMM: C = A @ B
// A [Nrows,K] row-major; B [K,FP] row-major (FP = 128/192, zero padded);
// C [Nrows,FP] row-major. One wave = 16x64 tile, 4 accumulators, A reused 4x.
// FP is a template constant so B offsets are load-immediates and the only
// per-iteration address math is two pointer increments.
template <int FP>
__global__ __launch_bounds__(256) void gat_gemm_wmma(
        const float* __restrict__ A, const float* __restrict__ B,
        float* __restrict__ C, int Nrows, int K) {
    const int lane   = threadIdx.x & 31;
    const int wave   = threadIdx.x >> 5;
    const int laneLo = lane & 15;
    const int hi     = lane >> 4;                 // 0 or 1
    const int row0   = (blockIdx.y * 8 + wave) * 16;
    const int col0   = blockIdx.x * 64;
    const int rowC   = min(row0 + laneLo, Nrows - 1);   // clamp: loads always valid
    const int c0     = col0 + laneLo;

    v8f acc0 = {}, acc1 = {}, acc2 = {}, acc3 = {};

    const float* pA = A + (size_t)rowC * K + 2 * hi;          // this half's K pair
    const float* pB = B + (size_t)(2 * hi) * FP + c0;         // row (2*hi), this col

#pragma unroll 2
    for (int k0 = 0; k0 < K; k0 += 4) {
        v2f a;
        a.x = pA[0];
        a.y = pA[1];
        // load all four B pairs first (distinct regs -> loads clause + overlap)
        v2f b0v, b1v, b2v, b3v;
        b0v.x = pB[0];       b0v.y = pB[FP];
        b1v.x = pB[16];      b1v.y = pB[FP + 16];
        b2v.x = pB[32];      b2v.y = pB[FP + 32];
        b3v.x = pB[48];      b3v.y = pB[FP + 48];
        acc0 = __builtin_amdgcn_wmma_f32_16x16x4_f32(false, a, false, b0v, (short)0, acc0, false, false);
        acc1 = __builtin_amdgcn_wmma_f32_16x16x4_f32(false, a, false, b1v, (short)0, acc1, false, false);
        acc2 = __builtin_amdgcn_wmma_f32_16x16x4_f32(false, a, false, b2v, (short)0, acc2, false, false);
        acc3 = __builtin_amdgcn_wmma_f32_16x16x4_f32(false, a, false, b3v, (short)0, acc3, false, false);
        pA += 4;
        pB += 4 * FP;
    }

#pragma unroll
    for (int v = 0; v < 8; ++v) {
        int r = row0 + v + 8 * hi;                // C layout: VGPR v, half-wave hi
        if (r < Nrows) {
            float* Crow = C + (size_t)r * FP;
            Crow[c0]      = acc0[v];
            Crow[c0 + 16] = acc1[v];
            Crow[c0 + 32] = acc2[v];
            Crow[c0 + 48] = acc3[v];
        }
    }
}

// ------------------------------------------------- el = fs . al ; er = h @ vd
__global__ void gat_el_er(const float* __restrict__ fs, const float* __restrict__ al,
                          const float* __restrict__ hin, const float* __restrict__ vd,
                          float* __restrict__ el, float* __restrict__ er,
                          int ldfs, int Dh, int K) {
    int t = blockIdx.x * blockDim.x + threadIdx.x;
    if (t >= NN * HH) return;
    int n = t >> 2, h = t & 3;
    const float* fr = fs + (size_t)n * ldfs + h * Dh;
    const float* av = al + h * Dh;
    float s = 0.f;
    for (int d = 0; d < Dh; ++d) s += fr[d] * av[d];
    el[t] = s;
    const float* hr = hin + (size_t)n * K;
    float s2 = 0.f;
    for (int k = 0; k < K; ++k) s2 += hr[k] * vd[k * HH + h];
    er[t] = s2;
}

// ------------------------------------------------- edge pass A: segment max
__global__ __launch_bounds__(256) void gat_edge_max(
        const int* __restrict__ src, const int* __restrict__ dst,
        const float* __restrict__ el, const float* __restrict__ er,
        float* __restrict__ m) {
    long long t = (long long)blockIdx.x * blockDim.x + threadIdx.x;
    if (t >= (long long)EE * HH) return;
    long long k = t >> 2; int h = t & 3;
    int s = src[k], d = dst[k];
    float e = el[s * HH + h] + er[d * HH + h];
    e = (e > 0.f) ? e : 0.2f * e;                 // leaky relu
    atomMaxF(&m[d * HH + h], e);
}

// ------------------------------------------------- edge pass B: exp + denom
__global__ __launch_bounds__(256) void gat_edge_exp(
        const int* __restrict__ src, const int* __restrict__ dst,
        const float* __restrict__ el, const float* __restrict__ er,
        const float* __restrict__ m, float* __restrict__ ez,
        float* __restrict__ denom) {
    long long t = (long long)blockIdx.x * blockDim.x + threadIdx.x;
    if (t >= (long long)EE * HH) return;
    long long k = t >> 2; int h = t & 3;
    int s = src[k], d = dst[k];
    float e = el[s * HH + h] + er[d * HH + h];
    e = (e > 0.f) ? e : 0.2f * e;
    float z = __expf(e - m[d * HH + h]);
    ez[t] = z;
    atomAddF(&denom[d * HH + h], z);
}

// ------------------------------------------------- edge pass C: scatter msg
// one wave (32 lanes) per edge; float4 loads of fs[src], per-component alpha.
// Templated on DH so f/DH becomes shifts (32) or magic-mul (47).
template <int DH>
__global__ __launch_bounds__(256) void gat_edge_agg(
        const int* __restrict__ src, const int* __restrict__ dst,
        const float* __restrict__ fs, const float* __restrict__ ez,
        const float* __restrict__ denom, float* __restrict__ rst,
        int F, int ldfs) {
    int wave = threadIdx.x >> 5;
    int lane = threadIdx.x & 31;
    long long e = (long long)blockIdx.x * 8 + wave;
    if (e >= EE) return;
    int s = src[e], d = dst[e];
    const float* ezp = ez + e * HH;
    const float* dep = denom + (size_t)d * HH;
    float a0 = ezp[0] / dep[0];
    float a1 = ezp[1] / dep[1];
    float a2 = ezp[2] / dep[2];
    float a3 = ezp[3] / dep[3];
    const float4* fr4 = (const float4*)(fs + (size_t)s * ldfs);
    float*        rr  = rst + (size_t)d * F;
    const int nQ = F >> 2;                        // F is a multiple of 4 (128, 188)
    for (int q = lane; q < nQ; q += 32) {
        float4 v = fr4[q];
        int f = q * 4;
        int h0 = (f)     / DH, h1 = (f + 1) / DH;
        int h2 = (f + 2) / DH, h3 = (f + 3) / DH;
        float b0 = (h0 == 0) ? a0 : (h0 == 1) ? a1 : (h0 == 2) ? a2 : a3;
        float b1 = (h1 == 0) ? a0 : (h1 == 1) ? a1 : (h1 == 2) ? a2 : a3;
        float b2 = (h2 == 0) ? a0 : (h2 == 1) ? a1 : (h2 == 2) ? a2 : a3;
        float b3 = (h3 == 0) ? a0 : (h3 == 1) ? a1 : (h3 == 2) ? a2 : a3;
        atomAddF(&rr[f],     v.x * b0);
        atomAddF(&rr[f + 1], v.y * b1);
        atomAddF(&rr[f + 2], v.z * b2);
        atomAddF(&rr[f + 3], v.w * b3);
    }
}

// ------------------------------------------------- bias + relu (layers 0,1)
__global__ void gat_bias_relu(const float* __restrict__ rst, const float* __restrict__ b,
                              float* __restrict__ hout, long long NF, int F) {
    long long t = (long long)blockIdx.x * blockDim.x + threadIdx.x;
    if (t >= NF) return;
    int f = (int)(t % F);
    float v = rst[t] + b[f];
    hout[t] = (v > 0.f) ? v : 0.f;
}

// ------------------------------------------------- head-mean + log_softmax
__global__ void gat_final(const float* __restrict__ rst2, const float* __restrict__ b2,
                          float* __restrict__ out) {
    int wave = threadIdx.x >> 5;
    int lane = threadIdx.x & 31;
    int n = blockIdx.x * 8 + wave;
    if (n >= NN) return;
    const float* r = rst2 + (size_t)n * 188;      // [4,47] flat
    int c1 = lane, c2 = lane + 32;
    float v1 = 0.f, v2 = 0.f;
#pragma unroll
    for (int h = 0; h < 4; ++h) {
        v1 += r[h * 47 + c1] + b2[h * 47 + c1];
        if (c2 < 47) v2 += r[h * 47 + c2] + b2[h * 47 + c2];
    }
    v1 *= 0.25f; v2 *= 0.25f;
    float mx = (c2 < 47) ? fmaxf(v1, v2) : v1;
    for (int o = 16; o > 0; o >>= 1) mx = fmaxf(mx, __shfl_xor(mx, o, 32));
    float s = __expf(v1 - mx) + ((c2 < 47) ? __expf(v2 - mx) : 0.f);
    for (int o = 16; o > 0; o >>= 1) s += __shfl_xor(s, o, 32);
    float lse = __logf(s);
    out[(size_t)n * 47 + c1] = v1 - mx - lse;
    if (c2 < 47) out[(size_t)n * 47 + c2] = v2 - mx - lse;
}

// ================================================================ host side
static void run_gat_layer(const float* hin, int K,
                          const float* Ws, const float* Wd,
                          const float* al, const float* ar,
                          const int* src, const int* dst,
                          int F, int Dh,
                          float* fs, float* rst, float* el, float* er,
                          float* mbuf, float* denom, float* ezbuf, float* vd,
                          float* Bpad, hipStream_t stream) {
    const long long NH = (long long)NN * HH;
    const long long NF = (long long)NN * F;
    const long long EH = (long long)EE * HH;
    const int Fp = (F + 63) & ~63;                // 128 or 192

    gat_fill_f32<<<(unsigned)((NH + 255) / 256), 256, 0, stream>>>(mbuf, -3.0e38f, NH);
    gat_fill_f32<<<(unsigned)((NH + 255) / 256), 256, 0, stream>>>(denom, 0.f, NH);
    gat_fill_f32<<<(unsigned)((NF + 255) / 256), 256, 0, stream>>>(rst, 0.f, NF);

    gat_fold_vd<<<(unsigned)((K * HH + 255) / 256), 256, 0, stream>>>(Wd, ar, vd, K, F, Dh);

    const float* Bmat = Ws;
    if (Fp != F) {
        gat_pad_B<<<(unsigned)((K * Fp + 255) / 256), 256, 0, stream>>>(Ws, Bpad, K, F, Fp);
        Bmat = Bpad;
    }

    dim3 g(Fp / 64, (NN + 127) / 128);            // wave = 16x64 tile, 8 waves/block
    if (Fp == 128)
        gat_gemm_wmma<128><<<g, 256, 0, stream>>>(hin, Bmat, fs, NN, K);
    else
        gat_gemm_wmma<192><<<g, 256, 0, stream>>>(hin, Bmat, fs, NN, K);

    gat_el_er<<<(unsigned)((NH + 255) / 256), 256, 0, stream>>>(fs, al, hin, vd, el, er, Fp, Dh, K);

    gat_edge_max<<<(unsigned)((EH + 255) / 256), 256, 0, stream>>>(src, dst, el, er, mbuf);
    gat_edge_exp<<<(unsigned)((EH + 255) / 256), 256, 0, stream>>>(src, dst, el, er, mbuf, ezbuf, denom);
    if (Dh == 32)
        gat_edge_agg<32><<<(unsigned)((EE + 7) / 8), 256, 0, stream>>>(src, dst, fs, ezbuf, denom, rst, F, Fp);
    else
        gat_edge_agg<47><<<(unsigned)((EE + 7) / 8), 256, 0, stream>>>(src, dst, fs, ezbuf, denom, rst, F, Fp);
}

extern "C" void kernel_launch(void* const* d_in, const int* in_sizes, int n_in,
                              void* d_out, int out_size, void* d_ws, size_t ws_size,
                              hipStream_t stream) {
    const float* x   = (const float*)d_in[0];
    const int*   src = (const int*)  d_in[1];
    const int*   dst = (const int*)  d_in[2];
    const float* W0s = (const float*)d_in[3];
    const float* W0d = (const float*)d_in[4];
    const float* a0l = (const float*)d_in[5];
    const float* a0r = (const float*)d_in[6];
    const float* b0  = (const float*)d_in[7];
    const float* W1s = (const float*)d_in[8];
    const float* W1d = (const float*)d_in[9];
    const float* a1l = (const float*)d_in[10];
    const float* a1r = (const float*)d_in[11];
    const float* b1  = (const float*)d_in[12];
    const float* W2s = (const float*)d_in[13];
    const float* W2d = (const float*)d_in[14];
    const float* a2l = (const float*)d_in[15];
    const float* a2r = (const float*)d_in[16];
    const float* b2  = (const float*)d_in[17];
    float* out = (float*)d_out;

    float* ws = (float*)d_ws;
    size_t off = 0;
    float* fs    = ws + off; off += (size_t)NN * 192;   // fs (ld = Fp)
    float* rst   = ws + off; off += (size_t)NN * 192;   // aggregation target (ld = F)
    float* hbuf  = ws + off; off += (size_t)NN * 128;   // h1 / h2
    float* el    = ws + off; off += (size_t)NN * HH;
    float* er    = ws + off; off += (size_t)NN * HH;
    float* mbuf  = ws + off; off += (size_t)NN * HH;
    float* denom = ws + off; off += (size_t)NN * HH;
    float* ezbuf = ws + off; off += (size_t)EE * HH;
    float* vd    = ws + off; off += 256 * HH;
    float* Bpad  = ws + off; off += 256 * 192;          // padded weight matrix

    // ---- layer 0: 256 -> (4,32)
    run_gat_layer(x, 256, W0s, W0d, a0l, a0r, src, dst, 128, 32,
                  fs, rst, el, er, mbuf, denom, ezbuf, vd, Bpad, stream);
    {
        long long NF = (long long)NN * 128;
        gat_bias_relu<<<(unsigned)((NF + 255) / 256), 256, 0, stream>>>(rst, b0, hbuf, NF, 128);
    }

    // ---- layer 1: 128 -> (4,32)
    run_gat_layer(hbuf, 128, W1s, W1d, a1l, a1r, src, dst, 128, 32,
                  fs, rst, el, er, mbuf, denom, ezbuf, vd, Bpad, stream);
    {
        long long NF = (long long)NN * 128;
        gat_bias_relu<<<(unsigned)((NF + 255) / 256), 256, 0, stream>>>(rst, b1, hbuf, NF, 128);
    }

    // ---- layer 2: 128 -> (4,47), no activation
    run_gat_layer(hbuf, 128, W2s, W2d, a2l, a2r, src, dst, 188, 47,
                  fs, rst, el, er, mbuf, denom, ezbuf, vd, Bpad, stream);

    // ---- mean over heads + log_softmax -> out [N,47]
    gat_final<<<(unsigned)((NN + 7) / 8), 256, 0, stream>>>(rst, b2, out);
}